// QuantumLayer_23725399343671
// MI455X (gfx1250) — compile-verified
//
#include <hip/hip_runtime.h>

// QuantumLayer collapses analytically:
//   state before CNOTs is a product state (only 1q gates on |0000>).
//   Conjugating Z3 Z2 Z1 back through CX(3,0) CX(2,3) CX(1,2) CX(0,1)
//   gives Z3 Z1 Z0.  RZ leaves <Z> invariant, and for RX(w)RY(x)|0>:
//   <Z> = cos(w) * cos(x).
// => out[b] = cos(w0)cos(w1)cos(w3) * cos(x[b,0]) cos(x[b,1]) cos(x[b,3])
//
// Pure streaming kernel: 10 MB total traffic -> ~0.45 us at 23.3 TB/s HBM.
// No matrix ops remain, so no WMMA; optimize data movement instead:
// b128 NT loads/stores + global_prefetch_b8 + v_cos_f32.

typedef float vf4 __attribute__((ext_vector_type(4)));

__global__ __launch_bounds__(256) void qlayer_zexp_kernel(
    const float* __restrict__ x,   // [B,4] row-major
    const float* __restrict__ w,   // [4]
    float* __restrict__ out,       // [B]
    int B)
{
    // Shared-weight constant; uniform address -> scalar loads, 3 v_cos.
    const float C = __cosf(w[0]) * __cosf(w[1]) * __cosf(w[3]);

    const int t    = blockIdx.x * blockDim.x + threadIdx.x;
    const int base = t << 2;  // 4 samples per thread

    const vf4* __restrict__ xv = (const vf4*)x;

    // Stream-ahead prefetch (gfx1250 global_prefetch_b8), ~16 KB ahead.
    __builtin_prefetch((const void*)(xv + base + 1024), 0, 0);

    if (base + 3 < B) {
        // 4x global_load_b128, non-temporal (single-touch data).
        vf4 a = __builtin_nontemporal_load(xv + base + 0);
        vf4 b = __builtin_nontemporal_load(xv + base + 1);
        vf4 c = __builtin_nontemporal_load(xv + base + 2);
        vf4 d = __builtin_nontemporal_load(xv + base + 3);

        vf4 r;
        r.x = C * __cosf(a.x) * __cosf(a.y) * __cosf(a.w);
        r.y = C * __cosf(b.x) * __cosf(b.y) * __cosf(b.w);
        r.z = C * __cosf(c.x) * __cosf(c.y) * __cosf(c.w);
        r.w = C * __cosf(d.x) * __cosf(d.y) * __cosf(d.w);

        // One global_store_b128, non-temporal (base is a multiple of 4,
        // out is hipMalloc-aligned -> 16B aligned).
        __builtin_nontemporal_store(r, (vf4*)(out + base));
    } else {
        // Tail (B not a multiple of 4; not hit for B = 524288).
        for (int i = base; i < B; ++i) {
            vf4 a = __builtin_nontemporal_load(xv + i);
            out[i] = C * __cosf(a.x) * __cosf(a.y) * __cosf(a.w);
        }
    }
}

extern "C" void kernel_launch(void* const* d_in, const int* in_sizes, int n_in,
                              void* d_out, int out_size, void* d_ws, size_t ws_size,
                              hipStream_t stream) {
    const float* x = (const float*)d_in[0];   // [B,4] float32
    const float* w = (const float*)d_in[1];   // [4]   float32
    float* out     = (float*)d_out;           // [B,1] float32

    const int B       = in_sizes[0] / 4;      // 524288
    const int threads = (B + 3) / 4;          // 4 samples per thread
    const int block   = 256;                  // 8 wave32 waves per block
    const int grid    = (threads + block - 1) / block;

    qlayer_zexp_kernel<<<grid, block, 0, stream>>>(x, w, out, B);
}